// EMAVQ_24292335026190
// MI455X (gfx1250) — compile-verified
//
#include <hip/hip_runtime.h>

typedef __attribute__((ext_vector_type(2))) float v2f;
typedef __attribute__((ext_vector_type(8))) float v8f;
typedef __attribute__((ext_vector_type(4))) unsigned int u32x4;
typedef __attribute__((ext_vector_type(8))) int i32x8;
typedef __attribute__((ext_vector_type(4))) int i32x4;

#define KC   4096      // number of codes
#define DIMV 256       // vector dim
#define NR   65536     // number of z rows
#define DECAYF 0.99f
#define EPSF   1e-5f

#define TILE_STRIDE 260   // 256 + 4 pad -> conflict-free ds_load_b64 (matches TDM pad)

// ---------------------------------------------------------------------------
// 1) per-code squared norms  c2[k] = sum_d codebook[k][d]^2
// ---------------------------------------------------------------------------
__global__ void emavq_c2_kernel(const float* __restrict__ cb, float* __restrict__ c2) {
    int k = blockIdx.x * blockDim.x + threadIdx.x;
    if (k >= KC) return;
    const float4* row = (const float4*)(cb + (size_t)k * DIMV);
    float s = 0.f;
#pragma unroll
    for (int i = 0; i < DIMV / 4; ++i) {
        float4 v = row[i];
        s += v.x * v.x + v.y * v.y + v.z * v.z + v.w * v.w;
    }
    c2[k] = s;
}

// ---------------------------------------------------------------------------
// TDM: DMA one 16-code x 256-dim fp32 codebook tile into LDS.
// D# pad feature inserts 4 DWORDs after every 256 DWORDs -> LDS row stride 260,
// identical to the bank-conflict-free layout the WMMA B-fragment reads expect.
// ---------------------------------------------------------------------------
__device__ __forceinline__ void tdm_issue_tile(const float* gsrc, unsigned lds_addr) {
    const unsigned long long ga = (unsigned long long)(uintptr_t)gsrc;
    u32x4 g0;
    g0[0] = 1u;                                              // count=1 (valid), user mode
    g0[1] = lds_addr;                                        // LDS byte address
    g0[2] = (unsigned)(ga & 0xFFFFFFFFu);                    // global_addr[31:0]
    g0[3] = (unsigned)((ga >> 32) & 0x01FFFFFFu) | (2u << 30); // global_addr[56:32], type=2
    i32x8 g1;
    g1[0] = (int)((2u << 16)      // data_size = 4 bytes
                | (1u << 20)      // pad_enable
                | (7u << 22)      // pad_interval: every 256 DWORDs
                | (3u << 25));    // pad_amount: 4 DWORDs (16B)
    g1[1] = (int)(256u << 16);    // tensor_dim0[15:0] = 256 (bits 63:48)
    g1[2] = (int)(4096u << 16);   // tensor_dim0 hi = 0, tensor_dim1[15:0] = 4096
    g1[3] = (int)(256u << 16);    // tensor_dim1 hi = 0, tile_dim0 = 256
    g1[4] = 16;                   // tile_dim1 = 16, tile_dim2 = 0 (2-D)
    g1[5] = 256;                  // tensor_dim0_stride[31:0] = 256 elements
    g1[6] = 0;                    // stride hi / tensor_dim1_stride lo
    g1[7] = 0;
    i32x4 gz = {0, 0, 0, 0};      // groups 2/3 unused for <=2-D tiles
#if defined(__clang_major__) && (__clang_major__ >= 23)
    i32x8 gz8 = {0, 0, 0, 0, 0, 0, 0, 0};
    __builtin_amdgcn_tensor_load_to_lds(g0, g1, gz, gz, gz8, 0);
#else
    __builtin_amdgcn_tensor_load_to_lds(g0, g1, gz, gz, 0);
#endif
}

// ---------------------------------------------------------------------------
// 2) fused fp32-WMMA GEMM + row argmin
//    score(row,code) = |c|^2 - 2 * dot(z_row, c)   (|z|^2 constant per row)
//    4 waves / block, 16 rows / wave, A panel register-resident (128 VGPRs).
//    Codebook tiles streamed by the Tensor Data Mover, double-buffered in LDS.
// ---------------------------------------------------------------------------
__launch_bounds__(128, 1)
__global__ void emavq_dist_argmin_kernel(const float* __restrict__ z,
                                         const float* __restrict__ cb,
                                         const float* __restrict__ c2,
                                         int*   __restrict__ idx_out,
                                         float* __restrict__ idx_f_out) {
    __shared__ __align__(16) float btile[2][16 * TILE_STRIDE];

    const int lane  = threadIdx.x & 31;
    const int wave  = threadIdx.x >> 5;
    const int m     = lane & 15;   // row (A) / code column (B)
    const int khalf = lane >> 4;   // ISA 16x4 fp32 A layout: lanes>=16 hold K=2,3
    const int row_base = blockIdx.x * 64 + wave * 16;

    // Preload A fragments for all DIM=256: 64 chunks of K=4, v2f each.
    // Lane m, VGPR pair = z[row][4*kk + 2*khalf + {0,1}]  (ISA A-matrix 16x4 layout)
    v2f a[64];
    const float* zrow = z + (size_t)(row_base + m) * DIMV + 2 * khalf;
#pragma unroll
    for (int kk = 0; kk < 64; ++kk)
        a[kk] = *(const v2f*)(zrow + 4 * kk);

    float best[8];
    int   bidx[8];
#pragma unroll
    for (int r = 0; r < 8; ++r) { best[r] = 3.4e38f; bidx[r] = 0; }

    const unsigned lds0 = (unsigned)(uintptr_t)&btile[0][0];
    const unsigned lds1 = (unsigned)(uintptr_t)&btile[1][0];

    // prologue: kick off tile 0 DMA
    if (wave == 0) tdm_issue_tile(cb, lds0);

    for (int tile = 0; tile < KC / 16; ++tile) {
        // everyone done with buf[(tile+1)&1] from iteration tile-1
        __syncthreads();
        if (wave == 0) {
            if (tile + 1 < KC / 16) {
                tdm_issue_tile(cb + (size_t)(tile + 1) * 16 * DIMV,
                               ((tile + 1) & 1) ? lds1 : lds0);
                // TDM ops from one wave complete in order: <=1 outstanding
                // means tile `tile` is fully resident in LDS.
                __builtin_amdgcn_s_wait_tensorcnt(1);
            } else {
                __builtin_amdgcn_s_wait_tensorcnt(0);
            }
        }
        __syncthreads();  // buf[tile&1] ready for all waves

        const float* buf = (tile & 1) ? &btile[1][0] : &btile[0][0];

        // 16x16 dot tile over K(dim)=256: 64 x v_wmma_f32_16x16x4_f32
        v8f acc = {};
#pragma unroll
        for (int kk = 0; kk < 64; ++kk) {
            // B fragment (4x16): lane holds B[2*khalf + {0,1}][m]
            //                  = cb_tile[code m][4*kk + 2*khalf + {0,1}]
            v2f b = *(const v2f*)&buf[m * TILE_STRIDE + kk * 4 + 2 * khalf];
            acc = __builtin_amdgcn_wmma_f32_16x16x4_f32(
                false, a[kk], false, b, (short)0, acc, false, false);
        }

        // epilogue: lane element r corresponds to (row = r + 8*khalf, code = tile*16 + m)
        const int   code0 = tile * 16 + m;
        const float c2v   = c2[code0];
#pragma unroll
        for (int r = 0; r < 8; ++r) {
            float score = __builtin_fmaf(-2.0f, acc[r], c2v);
            if (score < best[r]) { best[r] = score; bidx[r] = code0; }
        }
    }

    // cross-lane argmin over the 16 columns (stays within each 16-lane half)
#pragma unroll
    for (int r = 0; r < 8; ++r) {
        float s  = best[r];
        int   bi = bidx[r];
#pragma unroll
        for (int off = 1; off < 16; off <<= 1) {
            float so = __shfl_xor(s, off, 32);
            int   io = __shfl_xor(bi, off, 32);
            if (so < s || (so == s && io < bi)) { s = so; bi = io; }
        }
        best[r] = s; bidx[r] = bi;
    }

    if (m == 0) {  // lane 0 -> rows 0..7, lane 16 -> rows 8..15
#pragma unroll
        for (int r = 0; r < 8; ++r) {
            int row = row_base + r + 8 * khalf;
            idx_out[row]   = bidx[r];
            idx_f_out[row] = (float)bidx[r];  // indices slot of fp32 output buffer
        }
    }
}

// ---------------------------------------------------------------------------
// 3) gather z_q and segment-sum counts / sums via fp32 global atomics
// ---------------------------------------------------------------------------
__global__ void emavq_scatter_kernel(const float* __restrict__ z,
                                     const float* __restrict__ cb,
                                     const int*   __restrict__ idx,
                                     float* __restrict__ zq_out,
                                     float* __restrict__ counts,
                                     float* __restrict__ sums) {
    const int n = blockIdx.x;
    const int t = threadIdx.x;               // 64 threads x float4
    const int code = idx[n];
    const float4 zv = ((const float4*)(z + (size_t)n * DIMV))[t];
    ((float4*)(zq_out + (size_t)n * DIMV))[t] =
        ((const float4*)(cb + (size_t)code * DIMV))[t];
    float* s = sums + (size_t)code * DIMV + t * 4;
    atomicAdd(s + 0, zv.x);
    atomicAdd(s + 1, zv.y);
    atomicAdd(s + 2, zv.z);
    atomicAdd(s + 3, zv.w);
    if (t == 0) atomicAdd(counts + code, 1.0f);
}

// ---------------------------------------------------------------------------
// 4) EMA counts, total n, Laplace-smoothed denominators (single block)
// ---------------------------------------------------------------------------
__global__ void emavq_count_kernel(const float* __restrict__ ema_count,
                                   const float* __restrict__ counts,
                                   float* __restrict__ new_count_out,
                                   float* __restrict__ count_den) {
    __shared__ float red[256];
    const int t = threadIdx.x;
    float nc[16];
    float local = 0.f;
#pragma unroll
    for (int i = 0; i < 16; ++i) {
        int k = t * 16 + i;
        nc[i] = DECAYF * ema_count[k] + (1.0f - DECAYF) * counts[k];
        local += nc[i];
    }
    red[t] = local;
    __syncthreads();
    for (int s = 128; s > 0; s >>= 1) {
        if (t < s) red[t] += red[t + s];
        __syncthreads();
    }
    const float n = red[0];
    const float denom = n + (float)KC * EPSF;
#pragma unroll
    for (int i = 0; i < 16; ++i) {
        int k = t * 16 + i;
        new_count_out[k] = nc[i];
        count_den[k] = (nc[i] + EPSF) / denom * n;
    }
}

// ---------------------------------------------------------------------------
// 5) EMA weights + normalized codebook (pure streaming)
// ---------------------------------------------------------------------------
__global__ void emavq_finalize_kernel(const float* __restrict__ ema_weight,
                                      const float* __restrict__ sums,
                                      const float* __restrict__ count_den,
                                      float* __restrict__ new_cb_out,
                                      float* __restrict__ new_ew_out) {
    size_t i = (size_t)blockIdx.x * blockDim.x + threadIdx.x;
    int k = (int)(i >> 8);  // DIM = 256
    float ew = DECAYF * ema_weight[i] + (1.0f - DECAYF) * sums[i];
    new_ew_out[i] = ew;
    new_cb_out[i] = ew / count_den[k];
}

// ---------------------------------------------------------------------------
extern "C" void kernel_launch(void* const* d_in, const int* in_sizes, int n_in,
                              void* d_out, int out_size, void* d_ws, size_t ws_size,
                              hipStream_t stream) {
    const float* z          = (const float*)d_in[0];
    const float* codebook   = (const float*)d_in[1];
    const float* ema_count  = (const float*)d_in[2];
    const float* ema_weight = (const float*)d_in[3];

    float* out = (float*)d_out;
    // output tuple layout (flat, return order, fp32):
    const size_t OFF_ZQ  = 0;
    const size_t OFF_IDX = OFF_ZQ + (size_t)NR * DIMV;     // 16777216
    const size_t OFF_CB  = OFF_IDX + NR;                   // +65536
    const size_t OFF_CNT = OFF_CB + (size_t)KC * DIMV;     // +1048576
    const size_t OFF_EW  = OFF_CNT + KC;                   // +4096

    // workspace layout
    float* ws_c2     = (float*)d_ws;           // KC
    float* ws_counts = ws_c2 + KC;             // KC   (zeroed each call)
    float* ws_den    = ws_counts + KC;         // KC
    float* ws_sums   = ws_den + KC;            // KC*DIMV (zeroed each call)
    int*   ws_idx    = (int*)(ws_sums + (size_t)KC * DIMV);  // NR

    // zero counts..sums every call (graph-capture-safe, deterministic)
    hipMemsetAsync(ws_counts, 0, (size_t)(2 * KC + (size_t)KC * DIMV) * sizeof(float), stream);

    emavq_c2_kernel<<<KC / 256, 256, 0, stream>>>(codebook, ws_c2);

    emavq_dist_argmin_kernel<<<NR / 64, 128, 0, stream>>>(
        z, codebook, ws_c2, ws_idx, out + OFF_IDX);

    emavq_scatter_kernel<<<NR, 64, 0, stream>>>(
        z, codebook, ws_idx, out + OFF_ZQ, ws_counts, ws_sums);

    emavq_count_kernel<<<1, 256, 0, stream>>>(
        ema_count, ws_counts, out + OFF_CNT, ws_den);

    emavq_finalize_kernel<<<((size_t)KC * DIMV) / 256, 256, 0, stream>>>(
        ema_weight, ws_sums, ws_den, out + OFF_CB, out + OFF_EW);
}